// MatryoshkaQuantizer_41704132444222
// MI455X (gfx1250) — compile-verified
//
#include <hip/hip_runtime.h>

typedef __attribute__((ext_vector_type(2))) float v2f;
typedef __attribute__((ext_vector_type(8))) float v8f;

namespace {
constexpr int KF  = 24;    // fused feature dim: 6 (L0) + 6 (L1) + 12 (L2)
constexpr int TPB = 128;   // tokens per block (M)
constexpr int SWP = 1040;  // LDS pitch (floats) for weight rows: 2*SWP % 64 == 32 -> conflict-free halves
constexpr int SFP = 25;    // LDS pitch for feature rows (odd -> conflict-free column reads)
constexpr int NTH = 256;   // 8 waves (wave32)
}

// out[t, d] = sum_k f[t,k] * Wc[d,k], Wc = [g0*W0 | g1*W1 | g2*W2] (1024 x 24)
// computed as 16x16 tiles via chained V_WMMA_F32_16X16X4_F32 (6 x K=4).
__global__ __launch_bounds__(NTH) void matryoshka_wmma_kernel(
    const float* __restrict__ x,
    const float* __restrict__ x_ref,
    const float* __restrict__ W0,
    const float* __restrict__ W1,
    const float* __restrict__ W2,
    const float* __restrict__ gates,
    float* __restrict__ out,
    int D)                                   // 1024
{
    extern __shared__ float smem[];
    float* sW = smem;                        // [KF][SWP]  B matrix (K x N)
    float* sF = smem + KF * SWP;             // [TPB][SFP] A matrix (M x K), gates folded in

    const int tid = threadIdx.x;

    // ---- stage combined weight matrix Wc (KF x D) into LDS, transposed ----
    for (int i = tid; i < D * 6; i += NTH) {
        int d = i / 6, j = i - d * 6;
        sW[j * SWP + d] = W0[i];
    }
    for (int i = tid; i < D * 6; i += NTH) {
        int d = i / 6, j = i - d * 6;
        sW[(6 + j) * SWP + d] = W1[i];
    }
    for (int i = tid; i < D * 12; i += NTH) {
        int d = i / 12, j = i - d * 12;
        sW[(12 + j) * SWP + d] = W2[i];
    }

    // ---- per-token features: exact closed form of the hypercube softmaxes ----
    const float g0 = 1.f / (1.f + expf(-gates[0]));
    const float g1 = 1.f / (1.f + expf(-gates[1]));
    const float g2 = 1.f / (1.f + expf(-gates[2]));
    if (tid < TPB) {
        const long tok = (long)blockIdx.x * TPB + tid;
        const float INV = 2.0f / 0.3f;       // 2/TEMP
        float u[6], ur[6];
        float* f = &sF[tid * SFP];
        #pragma unroll
        for (int j = 0; j < 6; ++j) {
            float t  = tanhf(x    [tok * 6 + j] * INV);   // q0 = tanh(2x/T)
            float tr = tanhf(x_ref[tok * 6 + j] * INV);   // q_ref
            u [j] = tanhf(t  * INV);                      // tanh(2*q0/T)
            ur[j] = tanhf(tr * INV);                      // tanh(2*q_ref/T)
            f[j]  = g0 * t;                               // level-0 features
        }
        #pragma unroll
        for (int p = 0; p < 3; ++p) {
            f[6 + p] = g1 * 0.5f * (u[2*p] + u[2*p+1]);   // trit_p
            f[9 + p] = g1 * 0.5f * (u[2*p+1] - u[2*p]);   // dir_p
            f[12 + 4*p + 0] = g2 * u [2*p];               // hex feats
            f[12 + 4*p + 1] = g2 * u [2*p+1];
            f[12 + 4*p + 2] = g2 * ur[2*p];
            f[12 + 4*p + 3] = g2 * ur[2*p+1];
        }
    }
    __syncthreads();

    // ---- (TPB x 24) @ (24 x D) via V_WMMA_F32_16X16X4_F32 ----
    const int lane = tid & 31;
    const int w    = tid >> 5;    // wave -> M tile (16 tokens)
    const int hi   = lane >> 4;   // lane half selects K pair {0,1} vs {2,3}
    const int l15  = lane & 15;

    // A fragments: ISA f32 16x4 layout: VGPR r, lane L -> M = L&15, K = 2*(L>>4)+r
    v2f a[6];
    #pragma unroll
    for (int kb = 0; kb < 6; ++kb) {
        const float* fr = &sF[(w * 16 + l15) * SFP + kb * 4 + 2 * hi];
        a[kb].x = fr[0];
        a[kb].y = fr[1];
    }

    const int ntiles = D >> 4;
    const long orow0 = (long)blockIdx.x * TPB + w * 16 + 8 * hi;
    for (int nt = 0; nt < ntiles; ++nt) {
        // B fragments: VGPR r, lane L -> K = 2*(L>>4)+r, N = L&15
        v2f b[6];
        #pragma unroll
        for (int kb = 0; kb < 6; ++kb) {
            const int k0 = kb * 4 + 2 * hi;
            b[kb].x = sW[(k0    ) * SWP + nt * 16 + l15];
            b[kb].y = sW[(k0 + 1) * SWP + nt * 16 + l15];
        }
        v8f acc = {};
        #pragma unroll
        for (int kb = 0; kb < 6; ++kb) {
            acc = __builtin_amdgcn_wmma_f32_16x16x4_f32(
                false, a[kb], false, b[kb], (short)0, acc, false, false);
        }
        // C/D layout: VGPR r, lane L -> M = r + 8*(L>>4), N = L&15.
        // Output (256MB) > L2 (192MB): stream with non-temporal stores.
        float* o = out + orow0 * D + nt * 16 + l15;
        #pragma unroll
        for (int r = 0; r < 8; ++r) {
            __builtin_nontemporal_store(acc[r], o + (long)r * D);
        }
    }
}

extern "C" void kernel_launch(void* const* d_in, const int* in_sizes, int n_in,
                              void* d_out, int out_size, void* d_ws, size_t ws_size,
                              hipStream_t stream) {
    const float* x    = (const float*)d_in[0];
    const float* xr   = (const float*)d_in[1];
    const float* W0   = (const float*)d_in[2];
    const float* W1   = (const float*)d_in[3];
    const float* W2   = (const float*)d_in[4];
    const float* gate = (const float*)d_in[5];
    float* out = (float*)d_out;

    const int tokens = in_sizes[0] / 6;   // B*S = 65536
    const int D      = in_sizes[2] / 6;   // 1024
    const int grid   = tokens / TPB;      // 512 (exact for given shapes)
    const size_t shmem = (size_t)(KF * SWP + TPB * SFP) * sizeof(float); // ~110 KB < 320 KB/WGP

    matryoshka_wmma_kernel<<<grid, NTH, shmem, stream>>>(x, xr, W0, W1, W2, gate, out, D);
}